// GCNModel_48112223650413
// MI455X (gfx1250) — compile-verified
//
#include <hip/hip_runtime.h>

typedef __attribute__((ext_vector_type(2))) float v2f;
typedef __attribute__((ext_vector_type(8))) float v8f;

#define NN 10000
#define NE 320000

__device__ __forceinline__ float leakyf(float x) {
    return x >= 0.0f ? x : 0.01f * x;
}

// ---------------------------------------------------------------------------
// Generic fp32 WMMA GEMM: C[M,N] = post(A[M,K] @ B[K,N] (+ bias[N])), row-major.
// 256 threads = 8 wave32; each wave computes a 16(M) x 64(N) tile via
// V_WMMA_F32_16X16X4_F32; block tile = 128 x 64. K % 4 == 0, N % 64 == 0.
// ---------------------------------------------------------------------------
__global__ __launch_bounds__(256) void gemm_wmma_f32(
    const float* __restrict__ A, const float* __restrict__ B,
    const float* __restrict__ bias, float* __restrict__ C,
    int M, int K, int N, int applyLeaky)
{
    const int tid  = threadIdx.x;
    const int wave = tid >> 5;
    const int lane = tid & 31;
    const int half = lane >> 4;   // 0: lanes 0-15, 1: lanes 16-31
    const int l    = lane & 15;

    const int row0 = blockIdx.y * 128 + wave * 16;
    const int colb = blockIdx.x * 64;

    int arow = row0 + l;
    if (arow > M - 1) arow = M - 1;        // clamp; garbage rows never stored
    const float* Arow = A + (size_t)arow * K;

    v8f acc0 = {}, acc1 = {}, acc2 = {}, acc3 = {};

    for (int kk = 0; kk < K; kk += 4) {
        const int ka = kk + 2 * half;      // K=4 fragment: half-wave K split
        const float2 av = *(const float2*)(Arow + ka);
        v2f a; a.x = av.x; a.y = av.y;

        const float* Bk = B + (size_t)ka * N + colb + l;
        v2f b0, b1, b2, b3;
        b0.x = Bk[0];  b0.y = Bk[N + 0];
        b1.x = Bk[16]; b1.y = Bk[N + 16];
        b2.x = Bk[32]; b2.y = Bk[N + 32];
        b3.x = Bk[48]; b3.y = Bk[N + 48];

        acc0 = __builtin_amdgcn_wmma_f32_16x16x4_f32(false, a, false, b0, (short)0, acc0, false, false);
        acc1 = __builtin_amdgcn_wmma_f32_16x16x4_f32(false, a, false, b1, (short)0, acc1, false, false);
        acc2 = __builtin_amdgcn_wmma_f32_16x16x4_f32(false, a, false, b2, (short)0, acc2, false, false);
        acc3 = __builtin_amdgcn_wmma_f32_16x16x4_f32(false, a, false, b3, (short)0, acc3, false, false);
    }

    #pragma unroll
    for (int i = 0; i < 8; ++i) {
        const int r = row0 + 8 * half + i;   // C/D layout: vgpr i, half-wave M split
        if (r < M) {
            float v0 = acc0[i], v1 = acc1[i], v2 = acc2[i], v3 = acc3[i];
            if (bias) {
                v0 += bias[colb + 0  + l];
                v1 += bias[colb + 16 + l];
                v2 += bias[colb + 32 + l];
                v3 += bias[colb + 48 + l];
            }
            if (applyLeaky) {
                v0 = leakyf(v0); v1 = leakyf(v1); v2 = leakyf(v2); v3 = leakyf(v3);
            }
            float* Crow = C + (size_t)r * N + colb + l;
            Crow[0]  = v0;
            Crow[16] = v1;
            Crow[32] = v2;
            Crow[48] = v3;
        }
    }
}

// ---------------------------------------------------------------------------
// S[64,512] += z[k0:k1,:].T @ M6[k0:k1,:]   (split-K, atomic reduction)
// 128 threads = 4 waves; wave w owns M-tile rows [16w,16w+16); grid.x = N/64,
// grid.y = K chunks of 1000 (10 * 1000 == NN exactly).
// ---------------------------------------------------------------------------
#define KCHUNK 1000
__global__ __launch_bounds__(128) void gemm_zt_splitk(
    const float* __restrict__ Z,   // [NN, 64]
    const float* __restrict__ Mm,  // [NN, 512]
    float* __restrict__ S)         // [64, 512]
{
    const int tid  = threadIdx.x;
    const int wave = tid >> 5;
    const int lane = tid & 31;
    const int half = lane >> 4;
    const int l    = lane & 15;

    const int row0 = wave * 16;
    const int colb = blockIdx.x * 64;
    const int k0   = blockIdx.y * KCHUNK;
    const int k1   = k0 + KCHUNK;

    v8f acc0 = {}, acc1 = {}, acc2 = {}, acc3 = {};

    for (int kk = k0; kk < k1; kk += 4) {
        const int ka = kk + 2 * half;
        v2f a;                                  // A = z^T : A[m][k] = z[k][m]
        a.x = Z[(size_t)ka       * 64 + row0 + l];
        a.y = Z[(size_t)(ka + 1) * 64 + row0 + l];

        const float* Bk = Mm + (size_t)ka * 512 + colb + l;
        v2f b0, b1, b2, b3;
        b0.x = Bk[0];  b0.y = Bk[512 + 0];
        b1.x = Bk[16]; b1.y = Bk[512 + 16];
        b2.x = Bk[32]; b2.y = Bk[512 + 32];
        b3.x = Bk[48]; b3.y = Bk[512 + 48];

        acc0 = __builtin_amdgcn_wmma_f32_16x16x4_f32(false, a, false, b0, (short)0, acc0, false, false);
        acc1 = __builtin_amdgcn_wmma_f32_16x16x4_f32(false, a, false, b1, (short)0, acc1, false, false);
        acc2 = __builtin_amdgcn_wmma_f32_16x16x4_f32(false, a, false, b2, (short)0, acc2, false, false);
        acc3 = __builtin_amdgcn_wmma_f32_16x16x4_f32(false, a, false, b3, (short)0, acc3, false, false);
    }

    #pragma unroll
    for (int i = 0; i < 8; ++i) {
        const int r = row0 + 8 * half + i;      // always < 64
        float* Srow = S + (size_t)r * 512 + colb + l;
        unsafeAtomicAdd(Srow + 0,  acc0[i]);
        unsafeAtomicAdd(Srow + 16, acc1[i]);
        unsafeAtomicAdd(Srow + 32, acc2[i]);
        unsafeAtomicAdd(Srow + 48, acc3[i]);
    }
}

// ---------------------------------------------------------------------------
// SpMM scatter: H[rows[e], :] += vals[e] * Y[cols[e], :]   (256 feature cols)
// ---------------------------------------------------------------------------
#define EPB 16
__global__ __launch_bounds__(256) void spmm_scatter(
    const int* __restrict__ rows, const int* __restrict__ cols,
    const float* __restrict__ vals, const float* __restrict__ Y,
    float* __restrict__ H)
{
    const int t  = threadIdx.x;            // feature column 0..255
    const int e0 = blockIdx.x * EPB;
    #pragma unroll
    for (int i = 0; i < EPB; ++i) {
        const int e = e0 + i;
        if (e < NE) {
            const int   r = rows[e];
            const int   c = cols[e];
            const float v = vals[e];
            unsafeAtomicAdd(&H[(size_t)r * 256 + t], v * Y[(size_t)c * 256 + t]);
        }
    }
}

// H = leaky(H + b[col]) elementwise; ncolsMask = ncols-1 (power of two)
__global__ __launch_bounds__(256) void bias_leaky_kernel(
    float* __restrict__ H, const float* __restrict__ b, int total, int ncolsMask)
{
    const int idx = blockIdx.x * 256 + threadIdx.x;
    if (idx < total) {
        H[idx] = leakyf(H[idx] + b[idx & ncolsMask]);
    }
}

// ---------------------------------------------------------------------------
extern "C" void kernel_launch(void* const* d_in, const int* in_sizes, int n_in,
                              void* d_out, int out_size, void* d_ws, size_t ws_size,
                              hipStream_t stream) {
    const float* x    = (const float*)d_in[0];
    const int*   rows = (const int*)  d_in[1];
    const int*   cols = (const int*)  d_in[2];
    const float* vals = (const float*)d_in[3];
    const float* W1   = (const float*)d_in[4];
    const float* b1   = (const float*)d_in[5];
    const float* Wc1a = (const float*)d_in[6];
    const float* bc1a = (const float*)d_in[7];
    const float* Wc1b = (const float*)d_in[8];
    const float* bc1b = (const float*)d_in[9];
    const float* Wc2a = (const float*)d_in[10];
    const float* bc2a = (const float*)d_in[11];
    const float* Wc2b = (const float*)d_in[12];
    const float* bc2b = (const float*)d_in[13];
    const float* W6   = (const float*)d_in[14];
    const float* b6   = (const float*)d_in[15];

    float* out = (float*)d_out;                  // [NN, 512]
    float* z   = out + (size_t)NN * 512;         // [NN, 64]  (2nd tuple output)

    // workspace layout (floats), with liveness-based reuse (~41 MB):
    float* ws = (float*)d_ws;
    float* Y  = ws;                 // [NN,256]  2,560,000
    float* H  = ws + 2560000;       // [NN,256]  2,560,000
    float* C1 = ws;                 // [NN,128]  reuses Y (dead after spmm)
    float* C2 = ws + 1280000;       // [NN,128]
    float* h2 = ws + 2560000;       // [NN,256]  reuses H (dead after C1)
    float* M6 = ws + 5120000;       // [NN,512]  5,120,000
    float* S  = ws + 10240000;      // [64,512]  32,768

    const dim3 blk(256);
    const int  MB = (NN + 127) / 128;            // 79 row-blocks

    // 1) Y = x @ W1                       [NN,512]x[512,256]
    gemm_wmma_f32<<<dim3(4, MB), blk, 0, stream>>>(x, W1, nullptr, Y, NN, 512, 256, 0);

    // 2) H = spmm(adj, Y)
    hipMemsetAsync(H, 0, (size_t)NN * 256 * sizeof(float), stream);
    spmm_scatter<<<dim3(NE / EPB), blk, 0, stream>>>(rows, cols, vals, Y, H);

    // 3) H = leaky(H + b1)
    bias_leaky_kernel<<<dim3((NN * 256 + 255) / 256), blk, 0, stream>>>(H, b1, NN * 256, 255);

    // 4) C1 = leaky(H @ Wc1a + bc1a)      [NN,256]x[256,128]
    gemm_wmma_f32<<<dim3(2, MB), blk, 0, stream>>>(H, Wc1a, bc1a, C1, NN, 256, 128, 1);

    // 5) z = C1 @ Wc1b + bc1b             [NN,128]x[128,64]   -> d_out segment 2
    gemm_wmma_f32<<<dim3(1, MB), blk, 0, stream>>>(C1, Wc1b, bc1b, z, NN, 128, 64, 0);

    // 6) C2 = leaky(z @ Wc2a + bc2a)      [NN,64]x[64,128]
    gemm_wmma_f32<<<dim3(2, MB), blk, 0, stream>>>(z, Wc2a, bc2a, C2, NN, 64, 128, 1);

    // 7) h2 = leaky(C2 @ Wc2b + bc2b)     [NN,128]x[128,256]
    gemm_wmma_f32<<<dim3(4, MB), blk, 0, stream>>>(C2, Wc2b, bc2b, h2, NN, 128, 256, 1);

    // 8) M6 = h2 @ W6                     [NN,256]x[256,512]
    gemm_wmma_f32<<<dim3(8, MB), blk, 0, stream>>>(h2, W6, nullptr, M6, NN, 256, 512, 0);

    // 9) S = z.T @ M6                     [64,NN]x[NN,512] split-K + atomics
    hipMemsetAsync(S, 0, (size_t)64 * 512 * sizeof(float), stream);
    gemm_zt_splitk<<<dim3(8, NN / KCHUNK), dim3(128), 0, stream>>>(z, M6, S);

    // 10) out = z @ S + b6                [NN,64]x[64,512]
    //     == (z @ z.T) @ (h2 @ W6) + b6 by associativity — avoids the NxN A.
    gemm_wmma_f32<<<dim3(8, MB), blk, 0, stream>>>(z, S, b6, out, NN, 64, 512, 0);
}